// TS_GCN_54631984005612
// MI455X (gfx1250) — compile-verified
//
#include <hip/hip_runtime.h>
#include <hip/hip_bf16.h>
#include <math.h>

// ---------------------------------------------------------------------------
// TS-GCN distance-geometry refinement, fully fused: one workgroup per batch.
// B=512 batches, N=128 nodes. 256 threads = 8 wave32 per workgroup.
// D, C(=mask*W), and per-step S matrices live in LDS (~220 KB of the 320 KB
// WGP budget) so the 10 descent steps never touch HBM (total HBM traffic is
// the one-shot 170 MB read/write => ~7us at 23.3 TB/s; everything else is
// LDS + VALU + WMMA).
// The S(128x128) @ X(128x3) projection each step is a fully unrolled chain of
// 32 v_wmma_f32_16x16x4_f32 ops fed by immediate-offset ds_load_b64 pairs.
// ---------------------------------------------------------------------------

typedef __attribute__((ext_vector_type(2))) float v2f;
typedef __attribute__((ext_vector_type(8))) float v8f;

#define NN   128
#define SLD  132           // padded LDS row stride for S (breaks bank conflicts)
#define PLD  132           // padded row stride for P extraction buffer
#define TS   10
#define STEP_EPS 0.1f
#define ALPHA 5.0f
#define ALPHA_BASE 0.1f
#define KRANK 3
#define PSTEPS 10

__device__ __forceinline__ float softplus_f(float x) {
  // numerically stable softplus
  return fmaxf(x, 0.0f) + log1pf(expf(-fabsf(x)));
}

// Tree reduction over the 256-thread block; all threads must call it.
__device__ __forceinline__ float block_reduce(float v, float* red, int tid) {
  red[tid] = v;
  __syncthreads();
  #pragma unroll
  for (int s = 128; s > 0; s >>= 1) {
    if (tid < s) red[tid] += red[tid + s];
    __syncthreads();
  }
  float r = red[0];
  __syncthreads();
  return r;
}

__launch_bounds__(256, 1)
__global__ void ts_gcn_fused(const float* __restrict__ edge_pred,  // (B,N,N,2)
                             const int*   __restrict__ adj,        // (B,N,N)
                             const float* __restrict__ d_init,     // (1,)
                             const float* __restrict__ u_init,     // (B,N,3)
                             const float* __restrict__ x_noise,    // (B,N,3)
                             float*       __restrict__ out)        // (B,N,N)
{
  __shared__ float  Dp[NN * NN];       // 64 KB : masked softplus distances
  __shared__ float  Cw[NN * NN];       // 64 KB : mask*W weights
  __shared__ float  Sm[NN * SLD];      // 66 KB : Bg, then per-step S matrix
  __shared__ float2 Xq[64 * 16];       //  8 KB : WMMA B-layout X (zero-padded)
  __shared__ float  Pp[16 * PLD];      //  8 KB : WMMA D extraction (padded)
  __shared__ float  Xx[NN], Xy[NN], Xz[NN];
  __shared__ float  rsum[NN];
  __shared__ float  red[256];
  __shared__ float  ucol[KRANK][NN];
  __shared__ float  uvec[NN], utmp[NN];

  const int b   = blockIdx.x;
  const int tid = threadIdx.x;
  const float d0 = d_init[0];

  const float* ep = edge_pred + (size_t)b * NN * NN * 2;
  const int*   ad = adj       + (size_t)b * NN * NN;

  // ---------------- Phase 0: symmetrize + softplus -> Dp, Cw; Msum ----------
  float partM = 0.0f;
  for (int p = tid; p < NN * NN; p += 256) {
    const int i = p >> 7, j = p & 127;
    const float  m = (float)ad[p];
    const float2 e = ((const float2*)ep)[p];            // edge_pred[b,i,j,:]
    const float2 t = ((const float2*)ep)[j * NN + i];   // edge_pred[b,j,i,:]
    const float sp0 = softplus_f(d0 + e.x + t.x);
    const float sp1 = softplus_f(d0 + e.y + t.y);
    Dp[p] = sp0 * m;                 // D
    Cw[p] = sp1 * m;                 // mask*W == W (diag term is zero anyway)
    partM += (i == j) ? 1.0f : m;    // sum(mask)
  }
  // zero-init the padded B-fragment buffer (cols 3..15 stay zero forever)
  for (int idx = tid; idx < 64 * 16; idx += 256) Xq[idx] = make_float2(0.f, 0.f);
  const float Msum = block_reduce(partM, red, tid);

  // N_mol = sum_i mask[i,0]
  const float Nmol = block_reduce(
      (tid < NN) ? ((tid == 0) ? 1.0f : (float)ad[tid * NN]) : 0.0f, red, tid);

  // ---------------- Gram matrix Bg into Sm ----------------------------------
  {
    // row sums of D^2 (thread t handles half a row)
    const int i = tid >> 1, j0 = (tid & 1) << 6;
    float part = 0.0f;
    for (int jj = 0; jj < 64; ++jj) {
      const float d = Dp[i * NN + j0 + jj];
      part += d * d;
    }
    red[tid] = part;
    __syncthreads();
    if (tid < NN) rsum[tid] = red[2 * tid] + red[2 * tid + 1];
    __syncthreads();
    const float tot  = block_reduce((tid < NN) ? rsum[tid] : 0.0f, red, tid);
    const float invN = 1.0f / Nmol;
    const float Mn   = tot * invN * invN;
    for (int p = tid; p < NN * NN; p += 256) {
      const int i2 = p >> 7, j2 = p & 127;
      const float d  = Dp[p];
      const float d2 = d * d;
      const float mk = (i2 == j2) ? 1.0f : (d > 0.0f ? 1.0f : 0.0f);
      Sm[i2 * SLD + j2] =
          mk * -0.5f * (d2 - rsum[j2] * invN - rsum[i2] * invN + Mn);
    }
    __syncthreads();
  }

  // ---------------- Rank-3 power iteration with deflation -------------------
  for (int kx = 0; kx < KRANK; ++kx) {
    if (tid < NN) uvec[tid] = u_init[(size_t)b * NN * 3 + tid * 3 + kx];
    __syncthreads();
    for (int it = 0; it < PSTEPS; ++it) {
      const float nv =
          block_reduce((tid < NN) ? uvec[tid] * uvec[tid] : 0.0f, red, tid);
      const float inorm = 1.0f / fmaxf(sqrtf(nv), 0.001f);
      if (tid < NN) uvec[tid] *= inorm;
      __syncthreads();
      float dc0 = 0.0f, dc1 = 0.0f;
      if (kx > 0)
        dc0 = block_reduce((tid < NN) ? ucol[0][tid] * uvec[tid] : 0.0f, red, tid);
      if (kx > 1)
        dc1 = block_reduce((tid < NN) ? ucol[1][tid] * uvec[tid] : 0.0f, red, tid);
      if (tid < NN) {
        const float* row = &Sm[tid * SLD];
        float acc = 0.0f;
        #pragma unroll 8
        for (int j = 0; j < NN; ++j) acc = fmaf(row[j], uvec[j], acc);
        if (kx > 0) acc -= ucol[0][tid] * dc0;   // deflation: (A - u0 u0^T) u
        if (kx > 1) acc -= ucol[1][tid] * dc1;
        utmp[tid] = acc;
      }
      __syncthreads();
      if (tid < NN) uvec[tid] = utmp[tid];
      __syncthreads();
    }
    const float eig =
        block_reduce((tid < NN) ? uvec[tid] * uvec[tid] : 0.0f, red, tid);
    const float iden = 1.0f / sqrtf(sqrtf(eig + 0.01f));
    if (tid < NN) ucol[kx][tid] = uvec[tid] * iden;
    __syncthreads();
  }
  if (tid < NN) {
    const float* xn = x_noise + (size_t)b * NN * 3 + tid * 3;
    Xx[tid] = ucol[0][tid] + xn[0];
    Xy[tid] = ucol[1][tid] + xn[1];
    Xz[tid] = ucol[2][tid] + xn[2];
  }
  __syncthreads();
  // pack X into WMMA B-fragment layout: Xq[q*16+n] = {X[2q][n], X[2q+1][n]}
  if (tid < 192) {
    const int q = tid & 63, n = tid >> 6;                 // q in [0,64), n in [0,3)
    const float* comp = (n == 0) ? Xx : ((n == 1) ? Xy : Xz);
    Xq[q * 16 + n] = make_float2(comp[2 * q], comp[2 * q + 1]);
  }
  __syncthreads();

  // ---------------- T descent steps -----------------------------------------
  const float invM4 = 4.0f / Msum;
  const int lane   = tid & 31;
  const int mt     = tid >> 5;             // wave id -> M tile (8 waves, 8 tiles)
  const int nidx   = lane & 15;            // WMMA N index (xyz in 0..2)
  const int laneHi = lane >> 4;            // 0: K={0,1}, 1: K={2,3}
  const int arow   = mt * 16 + nidx;       // A-fragment row (M = lane&15)
  // per-lane fragment base addresses; loop offsets are compile-time immediates
  const float2* aq = (const float2*)&Sm[arow * SLD + laneHi * 2];  // +4 floats/kt
  const float2* bq = &Xq[laneHi * 16 + nidx];                      // +32 f2/kt

  for (int tstep = 0; tstep < TS; ++tstep) {
    // --- build S matrix + row sums: s_ij = (4/M) c_ij (Dx-D)/Dx -------------
    {
      const int i = tid >> 1, j0 = (tid & 1) << 6;
      const float xi0 = Xx[i], xi1 = Xy[i], xi2 = Xz[i];
      float part = 0.0f;
      for (int jj = 0; jj < 64; ++jj) {
        const int j = j0 + jj;
        const int p = i * NN + j;
        const float u0 = xi0 - Xx[j];
        const float u1 = xi1 - Xy[j];
        const float u2 = xi2 - Xz[j];
        const float dist = sqrtf(u0 * u0 + u1 * u1 + u2 * u2 + 0.01f);
        const float s = invM4 * Cw[p] * (dist - Dp[p]) / dist;
        Sm[i * SLD + j] = s;
        part += s;
      }
      red[tid] = part;
      __syncthreads();
      if (tid < NN) rsum[tid] = red[2 * tid] + red[2 * tid + 1];
      __syncthreads();
    }

    // --- P = S @ X : 32 chained v_wmma_f32_16x16x4_f32, fully unrolled ------
    // A fragment: 16x4 tile of S; B fragment: 4x16 tile of X^T (pre-packed).
    // Both feeds are single uniform ds_load_b64 with immediate offsets.
    v8f acc = {};
    #pragma unroll
    for (int kt = 0; kt < 32; ++kt) {
      const float2 apair = aq[kt * 2];      // Sm[arow*SLD + kt*4 + laneHi*2]
      const float2 bpair = bq[kt * 32];     // Xq[(kt*2+laneHi)*16 + nidx]
      v2f a;  a[0]  = apair.x;  a[1]  = apair.y;
      v2f bb; bb[0] = bpair.x;  bb[1] = bpair.y;
      acc = __builtin_amdgcn_wmma_f32_16x16x4_f32(
          /*neg_a=*/false, a, /*neg_b=*/false, bb,
          /*c_mod=*/(short)0, acc, /*reuse_a=*/false, /*reuse_b=*/false);
    }
    // D layout: VGPR r -> M = r (lanes0-15) / M = r+8 (lanes16-31), N = lane&15
    // Uniform padded stores (rows n>=3 are written but never read).
    {
      const int rbase = mt * 16 + laneHi * 8;
      #pragma unroll
      for (int r = 0; r < 8; ++r) Pp[nidx * PLD + rbase + r] = acc[r];
    }
    __syncthreads();

    // --- tanh-clipped gradient step -----------------------------------------
    if (tid < NN) {
      const float gx = rsum[tid] * Xx[tid] - Pp[0 * PLD + tid];
      const float gy = rsum[tid] * Xy[tid] - Pp[1 * PLD + tid];
      const float gz = rsum[tid] * Xz[tid] - Pp[2 * PLD + tid];
      const float dx = -STEP_EPS * gx;
      const float dy = -STEP_EPS * gy;
      const float dz = -STEP_EPS * gz;
      const float speed = sqrtf(dx * dx + dy * dy + dz * dz + 0.001f);
      const float at =
          ALPHA_BASE + (ALPHA - ALPHA_BASE) * ((float)(TS - tstep) / (float)TS);
      const float sc = at * tanhf(speed / at) / speed;
      Xx[tid] += dx * sc;
      Xy[tid] += dy * sc;
      Xz[tid] += dz * sc;
    }
    __syncthreads();
    // refresh packed B-fragment buffer with updated coordinates
    if (tid < 192) {
      const int q = tid & 63, n = tid >> 6;
      const float* comp = (n == 0) ? Xx : ((n == 1) ? Xy : Xz);
      Xq[q * 16 + n] = make_float2(comp[2 * q], comp[2 * q + 1]);
    }
    __syncthreads();
  }

  // ---------------- Output: adj * distances(X) ------------------------------
  float* ob = out + (size_t)b * NN * NN;
  for (int p = tid; p < NN * NN; p += 256) {
    const int i = p >> 7, j = p & 127;
    const float m  = (float)ad[p];
    const float u0 = Xx[i] - Xx[j];
    const float u1 = Xy[i] - Xy[j];
    const float u2 = Xz[i] - Xz[j];
    ob[p] = m * sqrtf(u0 * u0 + u1 * u1 + u2 * u2 + 0.01f);
  }
}

extern "C" void kernel_launch(void* const* d_in, const int* in_sizes, int n_in,
                              void* d_out, int out_size, void* d_ws, size_t ws_size,
                              hipStream_t stream) {
  const float* edge_pred = (const float*)d_in[0];
  const int*   adj       = (const int*)  d_in[1];
  const float* d_init    = (const float*)d_in[2];
  const float* u_init    = (const float*)d_in[3];
  const float* x_noise   = (const float*)d_in[4];
  float*       outp      = (float*)d_out;
  const int B = in_sizes[1] / (NN * NN);   // adj has B*N*N elements -> B=512
  ts_gcn_fused<<<B, 256, 0, stream>>>(edge_pred, adj, d_init, u_init, x_noise, outp);
}